// Model_91036126806563
// MI455X (gfx1250) — compile-verified
//
#include <hip/hip_runtime.h>

typedef float v2f __attribute__((ext_vector_type(2)));
typedef float v8f __attribute__((ext_vector_type(8)));

#define BATCH   2
#define SEQLEN  4096
#define NHEADS  32
#define DHEAD   64
#define NCHUNKS 16
#define CHUNK   256
#define DSTATE  128

// Pair-interleaved LDS layout: element (row s, col p) of a [S][64] matrix is at
// dword  (s>>1)*PP + ((2p + (s&1)) ^ (2*((s>>1)&15))).
// PP=160 == 32 mod 64  -> kh=0/kh=1 half-waves use disjoint bank halves.
// XOR swizzle is even and <32 -> preserves b64 pairs and 32-dword blocks,
// while decorrelating the pitch-strided staging stores from the banks.
#define PP      160
#define X_OFF   0                         // xw: 128 pair-rows
#define ST_OFF  (128 * PP)                // st: 64 pair-rows
#define W_OFF   (ST_OFF + 64 * PP)
#define EA_OFF  (W_OFF + CHUNK)
#define SMEM_FLOATS (EA_OFF + CHUNK)      // 31232 floats = 124,928 B

__device__ __forceinline__ int cswz(int pr) { return (pr & 15) << 1; }

__global__ __launch_bounds__(256)
void mamba_chunk_scan_kernel(const float* __restrict__ cb,
                             const float* __restrict__ x,
                             const float* __restrict__ dt,
                             const float* __restrict__ dA,
                             const float* __restrict__ Cm,
                             const float* __restrict__ prev,
                             const float* __restrict__ Dv,
                             float* __restrict__ out)
{
    extern __shared__ float smem[];
    float* xw_lds = smem + X_OFF;    // dt[s]*exp(-dA[s]) * x[s][p], pair-interleaved
    float* st_lds = smem + ST_OFF;   // prev_states^T [n][p], pair-interleaved
    float* w_lds  = smem + W_OFF;    // dt[s]*exp(-dA[s])  (staging only)
    float* ea_lds = smem + EA_OFF;   // exp(dA[l])

    const int bid = blockIdx.x;
    const int h   = bid & (NHEADS - 1);
    const int c   = (bid / NHEADS) & (NCHUNKS - 1);
    const int b   = bid / (NHEADS * NCHUNKS);
    const int tid = threadIdx.x;

    // ---------------- phase 1: per-position scalars ----------------
    {
        const size_t dbase = ((size_t)((b * NHEADS + h) * NCHUNKS + c)) * CHUNK;
        const float a = dA[dbase + tid];
        ea_lds[tid] = expf(a);
        w_lds[tid]  = dt[dbase + tid] * expf(-a);
    }
    __syncthreads();

    // ---------------- phase 2: matrix staging ----------------
    {
        // xw: task = (pair-row pr, p-pair pp). Two coalesced float2 row reads,
        // two b64 interleaved stores with the w[s] scale folded in.
        const size_t xbase = ((size_t)(b * SEQLEN + c * CHUNK) * NHEADS + h) * DHEAD;
        #pragma unroll
        for (int i = 0; i < 16; ++i) {
            const int t  = i * 256 + tid;
            const int pp = t & 31;           // p = 2*pp
            const int pr = t >> 5;           // s-pair 0..127
            const int s0 = pr << 1;
            const float2 ae = *(const float2*)(x + xbase + (size_t)s0 * (NHEADS * DHEAD) + 2 * pp);
            const float2 ao = *(const float2*)(x + xbase + (size_t)(s0 + 1) * (NHEADS * DHEAD) + 2 * pp);
            const float w0 = w_lds[s0];
            const float w1 = w_lds[s0 + 1];
            float* d = xw_lds + pr * PP;
            const int cx = cswz(pr);
            v2f v0; v0.x = ae.x * w0; v0.y = ao.x * w1;
            v2f v1; v1.x = ae.y * w0; v1.y = ao.y * w1;
            *(v2f*)(d + (((4 * pp)    ) ^ cx)) = v0;   // col p = 2pp
            *(v2f*)(d + (((4 * pp) + 2) ^ cx)) = v1;   // col p = 2pp+1
        }
        // st: task = (n-pair nq, row p). One coalesced float2 over n is exactly
        // one interleaved pair -> single b64 store.
        const size_t pbase = ((size_t)((b * NCHUNKS + c) * NHEADS + h)) * DHEAD * DSTATE;
        #pragma unroll
        for (int i = 0; i < 16; ++i) {
            const int t  = i * 256 + tid;
            const int nq = t & 63;           // n-pair 0..63
            const int p  = t >> 6;           // 0..63
            const float2 v = *(const float2*)(prev + pbase + (size_t)p * DSTATE + 2 * nq);
            v2f sv; sv.x = v.x; sv.y = v.y;
            *(v2f*)(st_lds + nq * PP + ((2 * p) ^ cswz(nq))) = sv;
        }
    }
    __syncthreads();

    const int wave = tid >> 5;
    const int lane = tid & 31;
    const int m    = lane & 15;   // A row / B-C-D column within tile
    const int kh   = lane >> 4;   // K-half (A/B); M-half (C/D)

    const size_t cb_base  = (size_t)(b * NCHUNKS + c) * CHUNK * CHUNK;
    const size_t C_base   = (size_t)(b * SEQLEN + c * CHUNK) * DSTATE;
    const size_t out_base = ((size_t)(b * SEQLEN + c * CHUNK) * NHEADS + h) * DHEAD;
    const float  Dh       = Dv[h];

    #pragma unroll
    for (int half = 0; half < 2; ++half) {
        const int lt = half ? (15 - wave) : wave;   // balanced causal split
        const int l  = lt * 16 + m;                 // this lane's A row

        v8f acc0 = {}; v8f acc1 = {}; v8f acc2 = {}; v8f acc3 = {};

        // ---- Part A bulk: strictly-below-diagonal K tiles, no masking ----
        const float* cbrow = cb + cb_base + (size_t)l * CHUNK;
        for (int kt = 0; kt < 4 * lt; ++kt) {
            const int s0 = kt * 4 + 2 * kh;          // A/B layout: K = 2*kh + j
            const v2f a = *(const v2f*)(cbrow + s0);
            const int pr = kt * 2 + kh;
            const float* base = xw_lds + pr * PP + (((m << 1)) ^ cswz(pr));
            const v2f b0 = *(const v2f*)(base +  0);
            const v2f b1 = *(const v2f*)(base + 32);
            const v2f b2 = *(const v2f*)(base + 64);
            const v2f b3 = *(const v2f*)(base + 96);
            acc0 = __builtin_amdgcn_wmma_f32_16x16x4_f32(false, a, false, b0, (short)0, acc0, false, false);
            acc1 = __builtin_amdgcn_wmma_f32_16x16x4_f32(false, a, false, b1, (short)0, acc1, false, false);
            acc2 = __builtin_amdgcn_wmma_f32_16x16x4_f32(false, a, false, b2, (short)0, acc2, false, false);
            acc3 = __builtin_amdgcn_wmma_f32_16x16x4_f32(false, a, false, b3, (short)0, acc3, false, false);
        }

        // ---- Part A diagonal tile: elementwise causal mask (branchless) ----
        for (int d = 0; d < 4; ++d) {
            const int kt = 4 * lt + d;
            const int s0 = kt * 4 + 2 * kh;
            const v2f cv = *(const v2f*)(cbrow + s0);
            v2f a;
            a.x = (s0     <= l) ? cv.x : 0.0f;       // value select, no guarded mem op
            a.y = (s0 + 1 <= l) ? cv.y : 0.0f;
            const int pr = kt * 2 + kh;
            const float* base = xw_lds + pr * PP + (((m << 1)) ^ cswz(pr));
            const v2f b0 = *(const v2f*)(base +  0);
            const v2f b1 = *(const v2f*)(base + 32);
            const v2f b2 = *(const v2f*)(base + 64);
            const v2f b3 = *(const v2f*)(base + 96);
            acc0 = __builtin_amdgcn_wmma_f32_16x16x4_f32(false, a, false, b0, (short)0, acc0, false, false);
            acc1 = __builtin_amdgcn_wmma_f32_16x16x4_f32(false, a, false, b1, (short)0, acc1, false, false);
            acc2 = __builtin_amdgcn_wmma_f32_16x16x4_f32(false, a, false, b2, (short)0, acc2, false, false);
            acc3 = __builtin_amdgcn_wmma_f32_16x16x4_f32(false, a, false, b3, (short)0, acc3, false, false);
        }

        // ---- Part B: inter-chunk  (C [256x128] x prev^T [128x64]) ----
        const float* Crow = Cm + C_base + (size_t)l * DSTATE;
        #pragma unroll 4
        for (int nk = 0; nk < DSTATE / 4; ++nk) {
            const int n0 = nk * 4 + 2 * kh;
            const v2f a = *(const v2f*)(Crow + n0);
            const int pr = nk * 2 + kh;
            const float* base = st_lds + pr * PP + (((m << 1)) ^ cswz(pr));
            const v2f b0 = *(const v2f*)(base +  0);
            const v2f b1 = *(const v2f*)(base + 32);
            const v2f b2 = *(const v2f*)(base + 64);
            const v2f b3 = *(const v2f*)(base + 96);
            acc0 = __builtin_amdgcn_wmma_f32_16x16x4_f32(false, a, false, b0, (short)0, acc0, false, false);
            acc1 = __builtin_amdgcn_wmma_f32_16x16x4_f32(false, a, false, b1, (short)0, acc1, false, false);
            acc2 = __builtin_amdgcn_wmma_f32_16x16x4_f32(false, a, false, b2, (short)0, acc2, false, false);
            acc3 = __builtin_amdgcn_wmma_f32_16x16x4_f32(false, a, false, b3, (short)0, acc3, false, false);
        }

        // ---- epilogue: row scale by exp(dA_l), add D*x skip (x from global) ----
        // C/D layout: VGPR r, lane = n + 16*mh -> row lt*16 + r + 8*mh, col pt*16 + n
        #pragma unroll
        for (int r = 0; r < 8; ++r) {
            const int   lo = lt * 16 + r + 8 * kh;
            const float e  = ea_lds[lo];
            const float* xr = x + out_base + (size_t)lo * (NHEADS * DHEAD) + m;
            float* orow = out + out_base + (size_t)lo * (NHEADS * DHEAD) + m;
            orow[0]  = acc0[r] * e + Dh * xr[0];
            orow[16] = acc1[r] * e + Dh * xr[16];
            orow[32] = acc2[r] * e + Dh * xr[32];
            orow[48] = acc3[r] * e + Dh * xr[48];
        }
    }
}

extern "C" void kernel_launch(void* const* d_in, const int* in_sizes, int n_in,
                              void* d_out, int out_size, void* d_ws, size_t ws_size,
                              hipStream_t stream) {
    const float* cb   = (const float*)d_in[0];
    const float* x    = (const float*)d_in[1];
    const float* dt   = (const float*)d_in[2];
    const float* dA   = (const float*)d_in[3];
    const float* Cm   = (const float*)d_in[4];
    const float* prev = (const float*)d_in[5];
    const float* Dv   = (const float*)d_in[6];
    float* out = (float*)d_out;

    const dim3 grid(BATCH * NCHUNKS * NHEADS);   // 1024 workgroups: (b, c, h)
    const dim3 block(256);                       // 8 wave32
    mamba_chunk_scan_kernel<<<grid, block, SMEM_FLOATS * sizeof(float), stream>>>(
        cb, x, dt, dA, Cm, prev, Dv, out);
}